// QLayerMultiQubit_40578851013037
// MI455X (gfx1250) — compile-verified
//
#include <hip/hip_runtime.h>

typedef __attribute__((ext_vector_type(2))) float v2f;
typedef __attribute__((ext_vector_type(8))) float v8f;

#define XSTRIDE 68   // padded LDS row stride (floats) for x tile: 16B-aligned float4 stores, bank-spread b64 reads
#define WSTRIDE 68
#define PSTRIDE 50   // padded stride for phi staging (48 cols)

__device__ __forceinline__ float2 cmul (float2 a, float2 b) { return float2{a.x*b.x - a.y*b.y, a.x*b.y + a.y*b.x}; }
__device__ __forceinline__ float2 cmulc(float2 a, float2 b) { return float2{a.x*b.x + a.y*b.y, a.y*b.x - a.x*b.y}; } // a * conj(b)
__device__ __forceinline__ float2 cadd (float2 a, float2 b) { return float2{a.x + b.x, a.y + b.y}; }

__global__ __launch_bounds__(256) void qlayer_kernel(
    const float* __restrict__ rho_re,
    const float* __restrict__ rho_im,
    const float* __restrict__ x,
    const float* __restrict__ w,
    const float* __restrict__ theta,
    float* __restrict__ out)
{
    __shared__ float x_lds[128 * XSTRIDE];   // 34816 B
    __shared__ float w_lds[48 * WSTRIDE];    // 13056 B
    __shared__ float phi_lds[8 * 16 * PSTRIDE]; // 25600 B

    const int tid  = threadIdx.x;
    const int lane = tid & 31;
    const int wv   = tid >> 5;
    const int m    = lane & 15;   // WMMA row/col within half
    const int hi   = lane >> 4;   // lane half selects K pair {0,1} vs {2,3}
    const long block_row0 = (long)blockIdx.x * 128;

    // ---- cooperative coalesced loads into LDS ----
    const float4* x4 = reinterpret_cast<const float4*>(x + block_row0 * 64);
#pragma unroll
    for (int it = 0; it < 8; ++it) {
        int q  = it * 256 + tid;        // 0..2047 float4s (128 rows x 16)
        int r  = q >> 4, c4 = q & 15;
        float4 v = x4[q];
        *reinterpret_cast<float4*>(&x_lds[r * XSTRIDE + c4 * 4]) = v;
    }
    const float4* w4 = reinterpret_cast<const float4*>(w);
#pragma unroll
    for (int it = 0; it < 3; ++it) {
        int q  = it * 256 + tid;        // 0..767 float4s (48 rows x 16)
        int r  = q >> 4, c4 = q & 15;
        float4 v = w4[q];
        *reinterpret_cast<float4*>(&w_lds[r * WSTRIDE + c4 * 4]) = v;
    }
    __syncthreads();

    // ---- WMMA f32 GEMM: each wave computes phi tile [16 rows x 48 cols] ----
    // acc[t] covers columns n = 16t .. 16t+15; init with theta[n] (bias folded in)
    v8f acc[3];
#pragma unroll
    for (int t = 0; t < 3; ++t) {
        float th = theta[t * 16 + m];
#pragma unroll
        for (int v = 0; v < 8; ++v) acc[t][v] = th;
    }
    const int arow = wv * 16 + m;       // local x row for this lane's A fragment
#pragma unroll
    for (int k0 = 0; k0 < 64; k0 += 4) {
        // A 16x4 f32 fragment: lane holds x[arow][k0 + 2*hi], x[arow][k0 + 2*hi + 1]
        v2f a = *reinterpret_cast<const v2f*>(&x_lds[arow * XSTRIDE + k0 + 2 * hi]);
#pragma unroll
        for (int t = 0; t < 3; ++t) {
            // B 4x16 f32 fragment: B[k][n] = w[n][k]; lane holds w[16t+m][k0+2*hi .. +1]
            v2f b = *reinterpret_cast<const v2f*>(&w_lds[(t * 16 + m) * WSTRIDE + k0 + 2 * hi]);
            acc[t] = __builtin_amdgcn_wmma_f32_16x16x4_f32(
                false, a, false, b, (short)0, acc[t], false, false);
        }
    }

    // ---- stage phi tiles to LDS (undo WMMA C layout) ----
    float* pw = &phi_lds[wv * 16 * PSTRIDE];
#pragma unroll
    for (int t = 0; t < 3; ++t)
#pragma unroll
        for (int v = 0; v < 8; ++v)
            pw[(v + 8 * hi) * PSTRIDE + t * 16 + m] = acc[t][v];
    __syncthreads();

    // ---- epilogue: build U, compute U rho U^dagger, coalesced rho/out traffic ----
#pragma unroll
    for (int it = 0; it < 8; ++it) {
        int p  = it * 256 + tid;        // 0..2047 pairs (128 rows x 16 c)
        int rl = p >> 4;                // local row
        int c  = p & 15;
        const float* ph = &phi_lds[(rl >> 4) * 16 * PSTRIDE + (rl & 15) * PSTRIDE + c * 3];
        float phi0 = ph[0], phi1 = ph[1], phi2 = ph[2];

        float sh, ch, sa, ca, sb, cb;
        __sincosf(0.5f * phi0,          &sh, &ch);
        __sincosf(0.5f * (phi1 + phi2), &sa, &ca);
        __sincosf(0.5f * (phi1 - phi2), &sb, &cb);

        float2 U00 = float2{ ch * ca,  ch * sa};
        float2 U01 = float2{-sh * cb, -sh * sb};
        float2 U10 = float2{ sh * cb, -sh * sb};
        float2 U11 = float2{ ch * ca, -ch * sa};

        long pair = (block_row0 + rl) * 16 + c;
        float4 rr = *reinterpret_cast<const float4*>(rho_re + pair * 4);
        float4 ri = *reinterpret_cast<const float4*>(rho_im + pair * 4);
        float2 r00 = float2{rr.x, ri.x}, r01 = float2{rr.y, ri.y};
        float2 r10 = float2{rr.z, ri.z}, r11 = float2{rr.w, ri.w};

        // T = U * rho
        float2 T00 = cadd(cmul(U00, r00), cmul(U01, r10));
        float2 T01 = cadd(cmul(U00, r01), cmul(U01, r11));
        float2 T10 = cadd(cmul(U10, r00), cmul(U11, r10));
        float2 T11 = cadd(cmul(U10, r01), cmul(U11, r11));

        // out[i][j] = sum_n T[i][n] * conj(U[j][n])
        float2 o00 = cadd(cmulc(T00, U00), cmulc(T01, U01));
        float2 o01 = cadd(cmulc(T00, U10), cmulc(T01, U11));
        float2 o10 = cadd(cmulc(T10, U00), cmulc(T11, U01));
        float2 o11 = cadd(cmulc(T10, U10), cmulc(T11, U11));

        float4* o = reinterpret_cast<float4*>(out + pair * 8);
        o[0] = float4{o00.x, o00.y, o01.x, o01.y};
        o[1] = float4{o10.x, o10.y, o11.x, o11.y};
    }
}

extern "C" void kernel_launch(void* const* d_in, const int* in_sizes, int n_in,
                              void* d_out, int out_size, void* d_ws, size_t ws_size,
                              hipStream_t stream) {
    const float* rho_re = (const float*)d_in[0];
    const float* rho_im = (const float*)d_in[1];
    const float* x      = (const float*)d_in[2];
    const float* w      = (const float*)d_in[3];
    const float* theta  = (const float*)d_in[4];
    float* out = (float*)d_out;

    const int B = in_sizes[2] / 64;          // 262144
    dim3 grid(B / 128), block(256);
    hipLaunchKernelGGL(qlayer_kernel, grid, block, 0, stream,
                       rho_re, rho_im, x, w, theta, out);
}